// nn_layer_23330262352162
// MI455X (gfx1250) — compile-verified
//
#include <hip/hip_runtime.h>

// Problem constants from the reference: b=2, s=8192, t=4096, e=128, NH=16
#define BB   2
#define SS   8192
#define TT   4096
#define EE   128
#define NH   16
#define STILE 16

typedef __attribute__((ext_vector_type(2))) float v2f;
typedef __attribute__((ext_vector_type(4))) float v4f;
typedef __attribute__((ext_vector_type(8))) float v8f;

// One wave (32 lanes) per tile of 16 targets.
// Distances computed as 16x16 tiles via V_WMMA_F32_16X16X4_F32:
//   D = C + A*B,  A = [tx ty 0 0] (16x4), B = [-2sx; -2sy; 0; 0] (4x16),
//   C[m][n] = |t_m|^2 + |s_n|^2  ->  D[m][n] = squared distance.
__global__ __launch_bounds__(32) void knn_wmma_kernel(
    const float* __restrict__ x,        // (B, S, E)
    const float* __restrict__ ct,       // (B, 2, T)
    const float* __restrict__ cs,       // (B, 2, S)
    float* __restrict__ out_x,          // (B, T, NH, E)
    long long* __restrict__ out_idx,    // (B, T, NH)
    float* __restrict__ out_cs)         // (B, 2, T, NH)
{
    __shared__ float tileD[16][17];     // padded: stride 17 -> conflict-free
    __shared__ int   sIdx[16][NH];

    const int lane = threadIdx.x;              // 0..31
    const int li   = lane & 15;                // mirrored half-wave index
    const int blk  = blockIdx.x;               // 0 .. B*(T/16)-1
    const int bb   = blk / (TT / 16);
    const int t0   = (blk % (TT / 16)) * 16;
    const int rowoff = (lane >= 16) ? 8 : 0;   // D-layout row offset for hi half

    const float* ctx = ct + (size_t)bb * 2 * TT;
    const float* cty = ctx + TT;
    const float* csx = cs + (size_t)bb * 2 * SS;
    const float* csy = csx + SS;

    // ---- A matrix: target coords (16x4 f32, K=2,3 zero) ----
    const float txr = ctx[t0 + li];
    const float tyr = cty[t0 + li];
    v2f a;
    a.x = (lane < 16) ? txr : 0.0f;            // K=0 on lanes 0-15, K=2 (=0) on hi
    a.y = (lane < 16) ? tyr : 0.0f;            // K=1 / K=3
    const float ntm = txr * txr + tyr * tyr;   // lane l (0-15) holds |t_{t0+l}|^2

    // Broadcast per-VGPR target norms needed for the C accumulator layout.
    float nt[8];
#pragma unroll
    for (int v = 0; v < 8; ++v)
        nt[v] = __shfl(ntm, v + rowoff, 32);

    // ---- per-lane sorted top-NH (lanes 0-15 each own one target row) ----
    float bd[NH];
    int   bi[NH];
#pragma unroll
    for (int k = 0; k < NH; ++k) { bd[k] = 3.0e38f; bi[k] = 0; }

    for (int s0 = 0; s0 < SS; s0 += STILE) {
        if (s0 + STILE < SS) {                 // gfx1250 global_prefetch_b8
            __builtin_prefetch(&csx[s0 + STILE + li], 0, 0);
            __builtin_prefetch(&csy[s0 + STILE + li], 0, 0);
        }

        // ---- B matrix: -2 * source coords (4x16 f32, K=2,3 zero) ----
        const float sxr = csx[s0 + li];
        const float syr = csy[s0 + li];
        v2f bm;
        bm.x = (lane < 16) ? (-2.0f * sxr) : 0.0f;
        bm.y = (lane < 16) ? (-2.0f * syr) : 0.0f;
        const float ns = sxr * sxr + syr * syr;   // per-lane |s_n|^2 (n = li)

        // ---- C accumulator: |t_m|^2 + |s_n|^2 in the 16x16 f32 D layout ----
        v8f c;
#pragma unroll
        for (int v = 0; v < 8; ++v) c[v] = nt[v] + ns;

        // D = A*B + C : one WMMA per 16x16 distance tile
        v8f d = __builtin_amdgcn_wmma_f32_16x16x4_f32(
            false, a, false, bm, (short)0, c, false, false);

        // Spill D tile to LDS: row m = v + rowoff, col n = li
#pragma unroll
        for (int v = 0; v < 8; ++v)
            tileD[v + rowoff][li] = d[v];
        __syncthreads();   // single-wave WG: S_NOP + compiler fence

        if (lane < 16) {
#pragma unroll
            for (int j = 0; j < STILE; ++j) {
                const float dd = tileD[lane][j];
                if (dd < bd[NH - 1]) {          // strict <: stable vs argsort
                    bd[NH - 1] = dd;
                    bi[NH - 1] = s0 + j;
#pragma unroll
                    for (int p = NH - 1; p > 0; --p) {
                        if (bd[p] < bd[p - 1]) {
                            float td = bd[p]; bd[p] = bd[p - 1]; bd[p - 1] = td;
                            int   ti = bi[p]; bi[p] = bi[p - 1]; bi[p - 1] = ti;
                        }
                    }
                }
            }
        }
        __syncthreads();
    }

    // ---- emit idx (int64) and coordinate diffs; stash idx for the gather ----
    if (lane < 16) {
        const int t = t0 + lane;
#pragma unroll
        for (int k = 0; k < NH; ++k) {
            const int si = bi[k];
            out_idx[((size_t)bb * TT + t) * NH + k] = (long long)si;
            out_cs[(((size_t)bb * 2 + 0) * TT + t) * NH + k] = txr - csx[si];
            out_cs[(((size_t)bb * 2 + 1) * TT + t) * NH + k] = tyr - csy[si];
            sIdx[lane][k] = si;
        }
    }
    __syncthreads();

    // ---- feature gather: 16 targets x 16 neighbors x 128 floats ----
    // One float4 per lane -> 32 lanes cover one 512B row per iteration.
    const float* xb = x + (size_t)bb * SS * EE;
    for (int p = 0; p < 16 * NH; ++p) {
        const int m  = p >> 4;
        const int k  = p & 15;
        const int si = sIdx[m][k];
        const v4f val = *(const v4f*)(xb + (size_t)si * EE + lane * 4);
        *(v4f*)(out_x + (((size_t)bb * TT + t0 + m) * NH + k) * (size_t)EE
                + lane * 4) = val;
    }
}

extern "C" void kernel_launch(void* const* d_in, const int* in_sizes, int n_in,
                              void* d_out, int out_size, void* d_ws, size_t ws_size,
                              hipStream_t stream) {
    (void)in_sizes; (void)n_in; (void)d_ws; (void)ws_size; (void)out_size;

    const float* x  = (const float*)d_in[0];   // (B, S, E) f32
    const float* ct = (const float*)d_in[1];   // (B, 2, T) f32
    const float* cs = (const float*)d_in[2];   // (B, 2, S) f32

    // Tuple outputs concatenated flat in return order: x_bs (f32), idx (i64), cs (f32)
    char* base = (char*)d_out;
    float*     out_x   = (float*)base;
    long long* out_idx = (long long*)(base + (size_t)BB * TT * NH * EE * sizeof(float));
    float*     out_cs  = (float*)(base + (size_t)BB * TT * NH * EE * sizeof(float)
                                       + (size_t)BB * TT * NH * sizeof(long long));

    dim3 grid(BB * (TT / 16));
    knn_wmma_kernel<<<grid, 32, 0, stream>>>(x, ct, cs, out_x, out_idx, out_cs);
}